// OpaqueScalar_20572893348677
// MI455X (gfx1250) — compile-verified
//
#include <hip/hip_runtime.h>

typedef __attribute__((ext_vector_type(2)))  float    v2f;
typedef __attribute__((ext_vector_type(4)))  float    v4f;
typedef __attribute__((ext_vector_type(8)))  float    v8f;

#define RED_BLOCKS 2048
#define RED_THREADS 256

__device__ __forceinline__ v8f wmma_rowsum(v2f a, v2f ones, v8f c) {
    // D[m][n] = sum_k a[m][k] * 1 + c : every element of `a` lands in the
    // accumulator replicated across the 16 columns (x16 total factor).
#if defined(__HIP_DEVICE_COMPILE__) && __has_builtin(__builtin_amdgcn_wmma_f32_16x16x4_f32)
    return __builtin_amdgcn_wmma_f32_16x16x4_f32(
        false, a, false, ones, (short)0, c, false, false);
#else
    // Host-pass parse stub only; never executed (kernels run on device, where
    // the real WMMA path above is compiled -- verified in device asm).
    c[0] += a.x * ones.x + a.y * ones.y;
    return c;
#endif
}

// Pass 1: stream x with NT b128 loads (data is 256MB, touched once -> bypass
// cache retention), accumulate via 4 independent WMMA chains, reduce to a
// per-block partial in d_ws.
__global__ void __launch_bounds__(RED_THREADS)
sum_reduce_wmma(const float* __restrict__ x, float* __restrict__ partial,
                long long n4, long long n) {
    const long long tid    = (long long)blockIdx.x * blockDim.x + threadIdx.x;
    const long long stride = (long long)gridDim.x * blockDim.x;
    // Uniform trip count across the whole grid -> EXEC all-ones inside the
    // WMMA loops (ISA requirement for V_WMMA).
    const long long trips  = n4 / stride;
    const long long nfull  = trips * stride;

    const v4f* __restrict__ xv = (const v4f*)x;
    const v2f ones2 = {1.0f, 1.0f};

    v8f acc0 = {}, acc1 = {}, acc2 = {}, acc3 = {};

    long long i = tid;
    long long t = 0;
    // Unroll x4: issue 4 independent b128 loads before consuming, so each
    // wave keeps 4 loads in flight (memory-level parallelism for HBM).
    for (; t + 4 <= trips; t += 4) {
        v4f d0 = __builtin_nontemporal_load(xv + i);
        v4f d1 = __builtin_nontemporal_load(xv + i + stride);
        v4f d2 = __builtin_nontemporal_load(xv + i + 2 * stride);
        v4f d3 = __builtin_nontemporal_load(xv + i + 3 * stride);
        i += 4 * stride;
        v2f a;
        a.x = d0.x; a.y = d0.y; acc0 = wmma_rowsum(a, ones2, acc0);
        a.x = d0.z; a.y = d0.w; acc1 = wmma_rowsum(a, ones2, acc1);
        a.x = d1.x; a.y = d1.y; acc2 = wmma_rowsum(a, ones2, acc2);
        a.x = d1.z; a.y = d1.w; acc3 = wmma_rowsum(a, ones2, acc3);
        a.x = d2.x; a.y = d2.y; acc0 = wmma_rowsum(a, ones2, acc0);
        a.x = d2.z; a.y = d2.w; acc1 = wmma_rowsum(a, ones2, acc1);
        a.x = d3.x; a.y = d3.y; acc2 = wmma_rowsum(a, ones2, acc2);
        a.x = d3.z; a.y = d3.w; acc3 = wmma_rowsum(a, ones2, acc3);
    }
    for (; t < trips; ++t) {
        v4f d = __builtin_nontemporal_load(xv + i);
        i += stride;
        v2f a;
        a.x = d.x; a.y = d.y; acc0 = wmma_rowsum(a, ones2, acc0);
        a.x = d.z; a.y = d.w; acc1 = wmma_rowsum(a, ones2, acc1);
    }

    float p = 0.0f;
    #pragma unroll
    for (int j = 0; j < 8; ++j) p += (acc0[j] + acc1[j]) + (acc2[j] + acc3[j]);

    // Ragged float4 remainder (scalar path; x16 to match the WMMA factor).
    {
        long long r = nfull + tid;
        if (r < n4) {
            v4f d = xv[r];
            p += 16.0f * ((d.x + d.y) + (d.z + d.w));
        }
    }
    // Sub-float4 tail.
    if (tid == 0) {
        for (long long k = n4 * 4; k < n; ++k) p += 16.0f * x[k];
    }

    __shared__ float sm[RED_THREADS];
    sm[threadIdx.x] = p;
    __syncthreads();
    for (int s = RED_THREADS / 2; s > 0; s >>= 1) {
        if ((int)threadIdx.x < s) sm[threadIdx.x] += sm[threadIdx.x + s];
        __syncthreads();
    }
    if (threadIdx.x == 0) partial[blockIdx.x] = sm[0];
}

// Pass 2: deterministic tree-reduce of block partials, scale by 1/16 * a * b.
__global__ void __launch_bounds__(RED_THREADS)
finalize(const float* __restrict__ partial, int nblocks,
         const float* __restrict__ a, const float* __restrict__ b,
         float* __restrict__ out) {
    float p = 0.0f;
    for (int i = threadIdx.x; i < nblocks; i += RED_THREADS) p += partial[i];
    __shared__ float sm[RED_THREADS];
    sm[threadIdx.x] = p;
    __syncthreads();
    for (int s = RED_THREADS / 2; s > 0; s >>= 1) {
        if ((int)threadIdx.x < s) sm[threadIdx.x] += sm[threadIdx.x + s];
        __syncthreads();
    }
    if (threadIdx.x == 0) out[0] = sm[0] * 0.0625f * a[0] * b[0];
}

extern "C" void kernel_launch(void* const* d_in, const int* in_sizes, int n_in,
                              void* d_out, int out_size, void* d_ws, size_t ws_size,
                              hipStream_t stream) {
    const float* x = (const float*)d_in[0];
    const float* a = (const float*)d_in[1];
    const float* b = (const float*)d_in[2];
    float* out     = (float*)d_out;
    float* partial = (float*)d_ws;   // RED_BLOCKS floats of scratch

    const long long n  = (long long)in_sizes[0];
    const long long n4 = n / 4;

    sum_reduce_wmma<<<RED_BLOCKS, RED_THREADS, 0, stream>>>(x, partial, n4, n);
    finalize<<<1, RED_THREADS, 0, stream>>>(partial, RED_BLOCKS, a, b, out);
}